// ESNPredictor_19653770347068
// MI455X (gfx1250) — compile-verified
//
#include <hip/hip_runtime.h>
#include <math.h>

typedef __attribute__((ext_vector_type(2))) float v2f;
typedef __attribute__((ext_vector_type(8))) float v8f;

#define B_DIM 128
#define T_DIM 64
#define F_DIM 256
#define R_DIM 1024
#define LEAK 0.3f
#define NOISE_SC 0.01f

#define NT 4        // 16-wide n-tiles per wave -> 64 columns per workgroup
#define KCHUNK 64   // K staged into LDS per pipeline stage
#define BK_PAD 66   // padded LDS row stride (floats) to dodge bank conflicts
#define NCHUNK_IN  (F_DIM / KCHUNK)              // 4  (x_t @ W_in)
#define NCHUNK_RES (R_DIM / KCHUNK)              // 16 (s @ W_res)
#define NCHUNKS    (NCHUNK_IN + NCHUNK_RES)      // 20
#define REGS_PER_THREAD 8   // 64x64 tile / 128 threads = 32 floats = 8 float4

// ---- software-pipeline helpers (128-thread workgroup) ----

// Issue global loads of a KCHUNK x 64 B tile (row-major, ld=R_DIM) into regs.
__device__ __forceinline__ void load_B_regs(const float* __restrict__ src,
                                            int kc, int n0, int tid,
                                            float4 r[REGS_PER_THREAD]) {
  const int kk0 = tid >> 4;         // 0..7
  const int nc  = (tid & 15) << 2;  // 0,4,...,60
#pragma unroll
  for (int i = 0; i < REGS_PER_THREAD; ++i) {
    r[i] = *(const float4*)(src + (size_t)(kc + i * 8 + kk0) * R_DIM + n0 + nc);
  }
}

// Write staged regs transposed into LDS as Blds[n][k] (so a WMMA B fragment,
// two consecutive K for one N, is a single contiguous float2 / ds_load_b64).
__device__ __forceinline__ void store_B_lds(float* __restrict__ Blds, int tid,
                                            const float4 r[REGS_PER_THREAD]) {
  const int kk0 = tid >> 4;
  const int nc  = (tid & 15) << 2;
#pragma unroll
  for (int i = 0; i < REGS_PER_THREAD; ++i) {
    const int kk = i * 8 + kk0;
    Blds[(nc + 0) * BK_PAD + kk] = r[i].x;
    Blds[(nc + 1) * BK_PAD + kk] = r[i].y;
    Blds[(nc + 2) * BK_PAD + kk] = r[i].z;
    Blds[(nc + 3) * BK_PAD + kk] = r[i].w;
  }
}

// KCHUNK of V_WMMA_F32_16X16X4_F32: A fragments (float2) from the lane's own
// global row, B fragments from transposed LDS; 4 independent accumulators per
// wave keep the matrix pipe issuing back-to-back.
__device__ __forceinline__ void wmma_kchunk(v8f acc[NT],
                                            const float* __restrict__ Arow,
                                            int kcA, int koff, int lane,
                                            const float* __restrict__ Blds) {
  const int nl = lane & 15;
#pragma unroll
  for (int k4 = 0; k4 < KCHUNK; k4 += 4) {
    const int kg = kcA + k4 + koff;
    v2f a;
    a.x = Arow[kg];
    a.y = Arow[kg + 1];
#pragma unroll
    for (int nt = 0; nt < NT; ++nt) {
      const float* bp = Blds + (nt * 16 + nl) * BK_PAD + k4 + koff;
      v2f b;
      b.x = bp[0];
      b.y = bp[1];
      acc[nt] = __builtin_amdgcn_wmma_f32_16x16x4_f32(
          false, a, false, b, (short)0, acc[nt], false, false);
    }
  }
}

// One fused ESN time step:
//   s_new = (1-LEAK)*s_old + LEAK*( tanh(x_t @ W_in + s_old @ W_res) + NOISE*n_t )
// Grid: (R_DIM/64 n-slices, 2 m-halves) = 32 workgroups of 128 threads
// (4 wave32s -> 1 wave per SIMD32: minimal XDL contention on the serial
// recurrent chain). Each wave owns 16 rows x 64 columns (4 f32 accumulators).
// LDS B tiles are double-buffered; next chunk's global loads are issued
// before the barrier so they retire under the current chunk's 64 WMMAs.
__global__ __launch_bounds__(128) void esn_step_kernel(
    const float* __restrict__ x, const float* __restrict__ W_in,
    const float* __restrict__ W_res, const float* __restrict__ noise,
    const float* __restrict__ s_old, float* __restrict__ s_new, int t) {
  __shared__ float Blds[2][64 * BK_PAD];

  const int tid  = threadIdx.x;
  const int lane = tid & 31;
  const int wave = tid >> 5;                 // 0..3
  const int n0   = blockIdx.x * 64;
  const int m0   = blockIdx.y * 64 + wave * 16;
  const int nl   = lane & 15;
  const int mrow = m0 + nl;                  // A-fragment row for this lane
  const int koff = (lane >> 4) << 1;         // K sub-offset per half-wave

  const float* Ax = x + ((size_t)mrow * T_DIM + t) * F_DIM;  // x[mrow, t, :]
  const float* As = s_old + (size_t)mrow * R_DIM;            // s_old[mrow, :]

  v8f acc[NT];
#pragma unroll
  for (int nt = 0; nt < NT; ++nt) acc[nt] = (v8f){};

  // chunk c: c < NCHUNK_IN -> (W_in, A = x_t); else -> (W_res, A = s_old)
  float4 regs[REGS_PER_THREAD];

  // prologue: stage chunk 0 into buffer 0
  load_B_regs(W_in, 0, n0, tid, regs);
  store_B_lds(Blds[0], tid, regs);

  for (int c = 0; c < NCHUNKS; ++c) {
    // issue next chunk's global loads before the barrier (overlap with WMMA)
    if (c + 1 < NCHUNKS) {
      const int cn = c + 1;
      const float* Bsrc = (cn < NCHUNK_IN) ? W_in : W_res;
      const int kcn = (cn < NCHUNK_IN) ? cn * KCHUNK
                                       : (cn - NCHUNK_IN) * KCHUNK;
      load_B_regs(Bsrc, kcn, n0, tid, regs);
    }

    __syncthreads();  // staged buffer (c & 1) is visible to all waves

    const float* Arow = (c < NCHUNK_IN) ? Ax : As;
    const int kcA = (c < NCHUNK_IN) ? c * KCHUNK : (c - NCHUNK_IN) * KCHUNK;
    wmma_kchunk(acc, Arow, kcA, koff, lane, Blds[c & 1]);

    // park next chunk in the other buffer; its prior readers finished
    // before the barrier above, so this is race-free
    if (c + 1 < NCHUNKS) {
      store_B_lds(Blds[(c + 1) & 1], tid, regs);
    }
  }

  // ---- epilogue: tanh, noise, leaky update ----
  // C layout: VGPR j holds M = j (lanes 0-15) / M = j+8 (lanes 16-31), N = lane%16.
  const int mlo = m0 + 8 * (lane >> 4);
#pragma unroll
  for (int nt = 0; nt < NT; ++nt) {
    const int n = n0 + nt * 16 + nl;
#pragma unroll
    for (int j = 0; j < 8; ++j) {
      const int m = mlo + j;
      const size_t idx = (size_t)m * R_DIM + n;
      const float pre =
          tanhf(acc[nt][j]) +
          NOISE_SC * noise[((size_t)t * B_DIM + m) * R_DIM + n];
      s_new[idx] = (1.0f - LEAK) * s_old[idx] + LEAK * pre;
    }
  }
}

// out = states @ W_out_w^T + W_out_b.  M=128, N=F_DIM, K=R_DIM.
// Both A and B are K-contiguous row-major here (B row = W_out_w[n, :]),
// so each lane's fragments are direct float2 loads; no LDS needed.
__global__ __launch_bounds__(256) void esn_out_kernel(
    const float* __restrict__ states, const float* __restrict__ W_out_w,
    const float* __restrict__ W_out_b, float* __restrict__ out) {
  const int tid  = threadIdx.x;
  const int lane = tid & 31;
  const int wave = tid >> 5;                 // 0..7 -> m-tile (covers 128 rows)
  const int n0   = blockIdx.x * 16;          // 16 output-feature columns
  const int nl   = lane & 15;
  const int mrow = wave * 16 + nl;
  const int koff = (lane >> 4) << 1;

  const float* Arow = states + (size_t)mrow * R_DIM;
  const float* Brow = W_out_w + (size_t)(n0 + nl) * R_DIM;

  v8f acc = {};
#pragma unroll 8
  for (int k = 0; k < R_DIM; k += 4) {
    v2f a, b;
    a.x = Arow[k + koff];
    a.y = Arow[k + koff + 1];
    b.x = Brow[k + koff];
    b.y = Brow[k + koff + 1];
    acc = __builtin_amdgcn_wmma_f32_16x16x4_f32(
        false, a, false, b, (short)0, acc, false, false);
  }

  const int n    = n0 + nl;
  const float bv = W_out_b[n];
  const int mlo  = wave * 16 + 8 * (lane >> 4);
#pragma unroll
  for (int j = 0; j < 8; ++j) {
    out[(size_t)(mlo + j) * F_DIM + n] = acc[j] + bv;
  }
}

__global__ void esn_zero_kernel(float* __restrict__ p, int n) {
  const int i = blockIdx.x * blockDim.x + threadIdx.x;
  if (i < n) p[i] = 0.0f;
}

extern "C" void kernel_launch(void* const* d_in, const int* in_sizes, int n_in,
                              void* d_out, int out_size, void* d_ws, size_t ws_size,
                              hipStream_t stream) {
  const float* x       = (const float*)d_in[0];  // [B, T, F]
  const float* W_in    = (const float*)d_in[1];  // [F, R]
  const float* W_res   = (const float*)d_in[2];  // [R, R]
  const float* W_out_w = (const float*)d_in[3];  // [F, R]
  const float* W_out_b = (const float*)d_in[4];  // [F]
  const float* noise   = (const float*)d_in[5];  // [T, B, R]

  float* s0 = (float*)d_ws;                 // ping-pong state buffers: 2 x 512 KB
  float* s1 = s0 + (size_t)B_DIM * R_DIM;

  const int nstate = B_DIM * R_DIM;
  esn_zero_kernel<<<(nstate + 255) / 256, 256, 0, stream>>>(s0, nstate);

  const float* s_in = s0;
  float* s_out = s1;
  dim3 step_grid(R_DIM / 64, B_DIM / 64);   // 16 x 2 = 32 workgroups
  for (int t = 0; t < T_DIM; ++t) {
    esn_step_kernel<<<step_grid, 128, 0, stream>>>(
        x, W_in, W_res, noise, s_in, s_out, t);
    float* tmp = (float*)s_in;
    s_in = s_out;
    s_out = tmp;
  }
  // T_DIM is even -> final states are back in s0 (== s_in here).
  esn_out_kernel<<<F_DIM / 16, 256, 0, stream>>>(
      s_in, W_out_w, W_out_b, (float*)d_out);
}